// SparseSimpleNeuralNetworkArchitectureZ_27573690040592
// MI455X (gfx1250) — compile-verified
//
#include <hip/hip_runtime.h>

typedef __attribute__((ext_vector_type(2))) float v2f;
typedef __attribute__((ext_vector_type(4))) float v4f;
typedef __attribute__((ext_vector_type(8))) float v8f;

#define D_IN 1536
#define HID 64
#define KP 64            // K-panel staged in LDS per outer iteration
#define ROWS 128         // rows of x per workgroup (8 waves x 16 rows)
#define XS_STRIDE 68     // 64 + 4 pad: conflict-free ds_load_b64 A-fragments
#define HS_STRIDE 68

// Weight panels live in LDS in k-quad-interleaved layout:
//   element (k, c) at  (k>>2)*256 + c*4 + (k&3)
// so the WMMA B-fragment (two consecutive k for one column) is one aligned
// ds_load_b64, and the two 16-lane halves (k0+0/1 vs k0+2/3) hit disjoint banks.

__launch_bounds__(256, 2)
__global__ void mlp3_fused_wmma(const float* __restrict__ x,
                                const float* __restrict__ W1,
                                const float* __restrict__ b1,
                                const float* __restrict__ W2,
                                const float* __restrict__ b2,
                                const float* __restrict__ W3,
                                const float* __restrict__ b3,
                                float* __restrict__ out) {
  __shared__ float xs[ROWS * XS_STRIDE];   // 34816 B
  __shared__ float w1q[KP * HID];          // 16384 B (quad-interleaved)
  __shared__ float w2q[HID * HID];         // 16384 B (quad-interleaved)
  __shared__ float hs[ROWS * HS_STRIDE];   // 34816 B  (total ~100 KB)

  const int tid  = threadIdx.x;
  const int lane = tid & 31;
  const int wave = tid >> 5;
  const int half = lane >> 4;   // which 16-lane half
  const int l16  = lane & 15;
  const long rowBase = (long)blockIdx.x * ROWS;

  // ---- Preload W2 (64x64) into LDS, quad-interleaved ----
  for (int i = tid; i < (HID * HID) / 4; i += 256) {
    int q = i >> 6;            // k-quad index
    int c = i & 63;            // column
    const float* src = W2 + (long)(q * 4) * HID + c;
    v4f v;
    v.x = src[0];
    v.y = src[HID];
    v.z = src[2 * HID];
    v.w = src[3 * HID];
    *(v4f*)(w2q + q * 256 + c * 4) = v;   // b128, conflict-light one-time store
  }

  // ---- Layer 1: acc[n] over K = 1536, N-tiles n = 0..3 ----
  v8f acc[4] = {};

  for (int kk = 0; kk < D_IN; kk += KP) {
    __syncthreads();
    // stage x tile: ROWS x KP, coalesced b128 loads, padded stride
    for (int i = tid; i < (ROWS * KP) / 4; i += 256) {
      int flat = i * 4;
      int r = flat >> 6;        // KP == 64
      int c = flat & 63;
      v4f v = *(const v4f*)(x + (rowBase + r) * D_IN + kk + c);
      *(v4f*)(xs + r * XS_STRIDE + c) = v;
    }
    // stage W1 panel: KP x 64, quad-interleaved
    for (int i = tid; i < (KP * HID) / 4; i += 256) {
      int q = i >> 6;
      int c = i & 63;
      const float* src = W1 + (long)(kk + q * 4) * HID + c;
      v4f v;
      v.x = src[0];
      v.y = src[HID];
      v.z = src[2 * HID];
      v.w = src[3 * HID];
      *(v4f*)(w1q + q * 256 + c * 4) = v;
    }
    __syncthreads();

    // prefetch next x panel into cache while we compute this one
    if (kk + KP < D_IN) {
      const float* pf = x + (rowBase + (tid >> 1)) * D_IN + kk + KP + (tid & 1) * 32;
      __builtin_prefetch(pf, 0, 3);
    }

    for (int k0 = 0; k0 < KP; k0 += 4) {
      const int kof = k0 + 2 * half;
      // A fragment: rows = wave*16 + l16, contiguous -> ds_load_b64
      v2f a = *(const v2f*)(xs + (wave * 16 + l16) * XS_STRIDE + kof);
      const int bbase = (k0 >> 2) * 256 + half * 2;
#pragma unroll
      for (int n = 0; n < 4; ++n) {
        // B fragment: single aligned ds_load_b64, no repacking movs
        v2f b = *(const v2f*)(w1q + bbase + (n * 16 + l16) * 4);
        acc[n] = __builtin_amdgcn_wmma_f32_16x16x4_f32(
            false, a, false, b, (short)0, acc[n], false, false);
      }
    }
  }

  // ---- bias + ReLU, store h1 to LDS (C-layout -> row-major relayout) ----
  float b1n[4];
#pragma unroll
  for (int n = 0; n < 4; ++n) b1n[n] = b1[n * 16 + l16];
#pragma unroll
  for (int n = 0; n < 4; ++n) {
#pragma unroll
    for (int j = 0; j < 8; ++j) {
      float v = fmaxf(acc[n][j] + b1n[n], 0.0f);
      hs[(wave * 16 + j + 8 * half) * HS_STRIDE + n * 16 + l16] = v;
    }
  }
  __syncthreads();

  // ---- Layer 2: h1[16x64] @ W2[64x64] per wave ----
  v8f c2[4] = {};
  for (int k0 = 0; k0 < HID; k0 += 4) {
    const int kof = k0 + 2 * half;
    v2f a = *(const v2f*)(hs + (wave * 16 + l16) * HS_STRIDE + kof);
    const int bbase = (k0 >> 2) * 256 + half * 2;
#pragma unroll
    for (int n = 0; n < 4; ++n) {
      v2f b = *(const v2f*)(w2q + bbase + (n * 16 + l16) * 4);
      c2[n] = __builtin_amdgcn_wmma_f32_16x16x4_f32(
          false, a, false, b, (short)0, c2[n], false, false);
    }
  }

  // ---- bias + ReLU on h2 (registers), then Layer 3 dot with W3 ----
  float b2n[4], w3n[4];
#pragma unroll
  for (int n = 0; n < 4; ++n) {
    b2n[n] = b2[n * 16 + l16];
    w3n[n] = W3[n * 16 + l16];   // W3 is [64,1]
  }
  float part[8];
#pragma unroll
  for (int j = 0; j < 8; ++j) {
    float s = 0.0f;
#pragma unroll
    for (int n = 0; n < 4; ++n) {
      float h2 = fmaxf(c2[n][j] + b2n[n], 0.0f);
      s += h2 * w3n[n];
    }
    part[j] = s;
  }
  // reduce across the 16 lanes of each half (masks < 16 keep halves separate)
#pragma unroll
  for (int j = 0; j < 8; ++j) {
    part[j] += __shfl_xor(part[j], 1, 32);
    part[j] += __shfl_xor(part[j], 2, 32);
    part[j] += __shfl_xor(part[j], 4, 32);
    part[j] += __shfl_xor(part[j], 8, 32);
  }
  const float bb3 = b3[0];
  if (l16 == 0) {
    long r0 = rowBase + wave * 16 + half * 8;   // half0 -> M=0..7, half1 -> M=8..15
#pragma unroll
    for (int j = 0; j < 8; ++j) {
      out[r0 + j] = fmaxf(part[j] + bb3, 0.0f);
    }
  }
}

extern "C" void kernel_launch(void* const* d_in, const int* in_sizes, int n_in,
                              void* d_out, int out_size, void* d_ws, size_t ws_size,
                              hipStream_t stream) {
  (void)in_sizes; (void)n_in; (void)d_ws; (void)ws_size; (void)out_size;
  const float* x  = (const float*)d_in[0];
  const float* W1 = (const float*)d_in[1];
  const float* b1 = (const float*)d_in[2];
  const float* W2 = (const float*)d_in[3];
  const float* b2 = (const float*)d_in[4];
  const float* W3 = (const float*)d_in[5];
  const float* b3 = (const float*)d_in[6];
  float* out = (float*)d_out;

  const int BATCH = 131072;
  dim3 grid(BATCH / ROWS), block(256);
  mlp3_fused_wmma<<<grid, block, 0, stream>>>(x, W1, b1, W2, b2, W3, b3, out);
}